// HybridKernel_65481071396147
// MI455X (gfx1250) — compile-verified
//
#include <hip/hip_runtime.h>

// CDNA5 / gfx1250, wave32.
typedef float v2f __attribute__((ext_vector_type(2)));
typedef float v8f __attribute__((ext_vector_type(8)));

// ---------------------------------------------------------------------------
// Phase 1: build the two 8192x16 state matrices in workspace.
//   u[row][ perm(i) ] = prod-state(a[row])[i]   (perm = CNOT-ladder = prefix XOR)
//   v[row][ i ]       = prod-state(b[row])[i]
// ---------------------------------------------------------------------------
__global__ __launch_bounds__(256) void qc_build_states(
    const float* __restrict__ a, const float* __restrict__ b,
    float* __restrict__ u, float* __restrict__ v, int n)
{
  int t = blockIdx.x * blockDim.x + threadIdx.x;
  if (t >= 2 * n) return;
  bool isB = t >= n;
  int row = isB ? (t - n) : t;
  const float* ang = (isB ? b : a) + (size_t)row * 4;

  float c[4], s[4];
#pragma unroll
  for (int w = 0; w < 4; ++w)
    __sincosf(0.5f * ang[w], &s[w], &c[w]);

  // 2-wire partial tensor products, then combine.
  float e01[4], e23[4];
#pragma unroll
  for (int i = 0; i < 4; ++i) {
    e01[i] = (((i >> 1) & 1) ? s[0] : c[0]) * ((i & 1) ? s[1] : c[1]);
    e23[i] = (((i >> 1) & 1) ? s[2] : c[2]) * ((i & 1) ? s[3] : c[3]);
  }

  float* dst = (isB ? v : u) + (size_t)row * 16;
#pragma unroll
  for (int i = 0; i < 16; ++i) {
    float p = e01[i >> 2] * e23[i & 3];
    // CNOT ladder permutation: bit p of j = parity(i >> p) (4-bit prefix XOR)
    int j = isB ? i : (i ^ (i >> 1) ^ (i >> 2) ^ (i >> 3));
    dst[j] = p;
  }
}

// ---------------------------------------------------------------------------
// Phase 2: out[n][m] = (sum_k u[n][k] * v[m][k])^2  via V_WMMA_F32_16X16X4_F32.
// K = 16 -> 4 chained WMMAs per 16x16 tile. Each wave computes a 16x64 strip
// so A fragments are loaded once per 4 tiles.
//
// f32 A (16x4) lane layout: lanes 0-15 hold K=0 (VGPR0), K=1 (VGPR1) for row
// M=lane; lanes 16-31 hold K=2,3.  f32 B (4x16): lanes 0-15 hold K=0,1 for
// col N=lane; lanes 16-31 hold K=2,3.  Both map to a contiguous float2 load
// at [row_or_col][kstep*4 + 2*(lane>>4)].
// f32 C/D (16x16): VGPR r -> M=r (lanes 0-15, N=lane) and M=r+8 (lanes 16-31).
// ---------------------------------------------------------------------------
__global__ __launch_bounds__(256) void qc_fidelity_wmma(
    const float* __restrict__ u, const float* __restrict__ v,
    float* __restrict__ out, int N, int M)
{
  const int lane = threadIdx.x & 31;
  const int wave = threadIdx.x >> 5;

  const int stripsPerRow = M >> 6;                 // 64-wide column strips
  const long strip = (long)blockIdx.x * 8 + wave;  // 8 waves per block
  const int n0 = (int)(strip / stripsPerRow) << 4;
  const int m0 = (int)(strip % stripsPerRow) << 6;
  if (n0 >= N) return;

  const int r  = lane & 15;         // row (A) / col (B) / N (D) within tile
  const int kh = (lane >> 4) << 1;  // K sub-offset: 0 (lanes 0-15) or 2

  // A fragments for rows n0..n0+15, all 16 K values (4 k-steps).
  const float* ua = u + (size_t)(n0 + r) * 16 + kh;
  v2f a0 = *(const v2f*)(ua + 0);
  v2f a1 = *(const v2f*)(ua + 4);
  v2f a2 = *(const v2f*)(ua + 8);
  v2f a3 = *(const v2f*)(ua + 12);

#pragma unroll
  for (int t = 0; t < 4; ++t) {
    const int mt = m0 + (t << 4);
    const float* vb = v + (size_t)(mt + r) * 16 + kh;
    v2f b0 = *(const v2f*)(vb + 0);
    v2f b1 = *(const v2f*)(vb + 4);
    v2f b2 = *(const v2f*)(vb + 8);
    v2f b3 = *(const v2f*)(vb + 12);

    v8f acc = {};
    acc = __builtin_amdgcn_wmma_f32_16x16x4_f32(false, a0, false, b0,
                                                (short)0, acc, false, false);
    acc = __builtin_amdgcn_wmma_f32_16x16x4_f32(false, a1, false, b1,
                                                (short)0, acc, false, false);
    acc = __builtin_amdgcn_wmma_f32_16x16x4_f32(false, a2, false, b2,
                                                (short)0, acc, false, false);
    acc = __builtin_amdgcn_wmma_f32_16x16x4_f32(false, a3, false, b3,
                                                (short)0, acc, false, false);

    acc = acc * acc;  // fidelity = amplitude^2

    const int rowBase = n0 + ((lane >> 4) << 3);
    float* op = out + (size_t)rowBase * M + (size_t)(mt + r);
#pragma unroll
    for (int q = 0; q < 8; ++q)
      op[(size_t)q * M] = acc[q];
  }
}

// ---------------------------------------------------------------------------
extern "C" void kernel_launch(void* const* d_in, const int* in_sizes, int n_in,
                              void* d_out, int out_size, void* d_ws, size_t ws_size,
                              hipStream_t stream) {
  const float* a = (const float*)d_in[0];
  const float* b = (const float*)d_in[1];
  const int N = in_sizes[0] / 4;  // 8192 rows of a
  const int M = in_sizes[1] / 4;  // 8192 rows of b
  float* out = (float*)d_out;

  float* u = (float*)d_ws;            // N x 16
  float* v = u + (size_t)N * 16;      // M x 16   (1 MB total for 8192/8192)

  {
    int threads = 2 * N;  // one thread per state row (a rows then b rows)
    int blocks = (threads + 255) / 256;
    qc_build_states<<<blocks, 256, 0, stream>>>(a, b, u, v, N);
  }
  {
    long strips = (long)(N / 16) * (M / 64);   // 16x64 strip per wave
    int blocks = (int)((strips + 7) / 8);      // 8 waves (256 threads) / block
    qc_fidelity_wmma<<<blocks, 256, 0, stream>>>(u, v, out, N, M);
  }
}